// GMP_34196529611440
// MI455X (gfx1250) — compile-verified
//
#include <hip/hip_runtime.h>

// GMP dims (match reference)
#define N_J     116   // 20 (a) + 48 (b) + 48 (c) complex coefficient/feature pairs
#define N_STEPS 58    // 232 real feature slots / 4 per V_WMMA_F32_16X16X4_F32

typedef __attribute__((ext_vector_type(2))) float v2f;
typedef __attribute__((ext_vector_type(8))) float v8f;

struct FD { int l; int pi; int k; };

// Feature decode: j -> (lag l for x, power-table index pi for |x| position, power k)
// positions span n-6..n+3 -> pi = offset + 6 in [0,9]
__host__ __device__ constexpr FD fdec(int j) {
  if (j < 20) {                       // a[k][l]: x[n-l] * |x[n-l]|^k
    int k = j / 4, l = j % 4;
    return FD{ l, 6 - l, k };
  }
  if (j < 68) {                       // b[k][l][m]: x[n-l] * |x[n-l-m]|^k
    int u = j - 20; int k = u / 12, r = u % 12, l = r / 4, m = r % 4;
    return FD{ l, 6 - (l + m), k };
  }
  int u = j - 68;                     // c[k][l][m]: x[n-l] * |x[n-l+m]|^k
  int k = u / 12, r = u % 12, l = r / 4, m = r % 4;
  return FD{ l, 6 - l + m, k };
}

template <int KB>
__device__ inline void gmp_step(const float (&pw)[10][5],
                                const float (&xvr)[4], const float (&xvi)[4],
                                const float2* ws, int half, int m, v8f& acc) {
  constexpr int j0 = 2 * KB, j1 = 2 * KB + 1;
  constexpr FD f0 = fdec(j0), f1 = fdec(j1);

  // This lane's complex coefficient (A-matrix rows 0/1; other rows zero)
  float2 w = ws[half ? j1 : j0];
  float a0 = (m == 0) ?  w.x : ((m == 1) ? w.y : 0.0f);   // slot re: (w_re, w_im)
  float a1 = (m == 0) ? -w.y : ((m == 1) ? w.x : 0.0f);   // slot im: (-w_im, w_re)

  // This lane's complex feature g_j = x[n-l] * |x[pos]|^k  (all indices constexpr)
  float P  = half ? pw[f1.pi][f1.k] : pw[f0.pi][f0.k];
  float xr = half ? xvr[f1.l] : xvr[f0.l];
  float xi = half ? xvi[f1.l] : xvi[f0.l];

  v2f aa; aa[0] = a0;     aa[1] = a1;
  v2f bb; bb[0] = xr * P; bb[1] = xi * P;

  // D = A(16x4) x B(4x16) + C, full f32. Accumulate across 58 steps.
  acc = __builtin_amdgcn_wmma_f32_16x16x4_f32(false, aa, false, bb,
                                              (short)0, acc, false, false);
}

template <int KB>
__device__ inline void gmp_chain(const float (&pw)[10][5],
                                 const float (&xvr)[4], const float (&xvi)[4],
                                 const float2* ws, int half, int m, v8f& acc) {
  if constexpr (KB < N_STEPS) {
    gmp_step<KB>(pw, xvr, xvi, ws, half, m, acc);
    gmp_chain<KB + 1>(pw, xvr, xvi, ws, half, m, acc);
  }
}

__global__ __launch_bounds__(256)
void gmp_wmma_kernel(const float* __restrict__ x_re, const float* __restrict__ x_im,
                     const float* __restrict__ a_re, const float* __restrict__ a_im,
                     const float* __restrict__ b_re, const float* __restrict__ b_im,
                     const float* __restrict__ c_re, const float* __restrict__ c_im,
                     float* __restrict__ out, int n_total) {
  // Stage all 116 complex coefficients into LDS (flat orders match numpy row-major)
  __shared__ float2 ws[N_J];
  int t = threadIdx.x;
  if (t < N_J) {
    float wre, wim;
    if (t < 20)      { wre = a_re[t];      wim = a_im[t]; }
    else if (t < 68) { wre = b_re[t - 20]; wim = b_im[t - 20]; }
    else             { wre = c_re[t - 68]; wim = c_im[t - 68]; }
    ws[t] = make_float2(wre, wim);
  }
  __syncthreads();

  int lane = t & 31;
  int wave = t >> 5;
  int half = lane >> 4;      // selects K pair (0,1) vs (2,3) within each WMMA
  int m    = lane & 15;      // A row index == B column (sample) index
  int n0   = blockIdx.x * 128 + wave * 16 + m;            // this lane's sample
  int n    = n0 < n_total ? n0 : n_total - 1;             // clamp, keep EXEC full

  // Per-lane: powers |x[n+d-6]|^k for d=0..9, k=0..4, plus x[n-l] for l=0..3
  float pw[10][5];
  float xvr[4], xvi[4];
#pragma unroll
  for (int d = 0; d < 10; ++d) {
    int q = n + d - 6;
    q = q < 0 ? 0 : (q > n_total - 1 ? n_total - 1 : q);  // jnp.clip semantics
    float re = x_re[q], im = x_im[q];
    float s2 = re * re + im * im;
    // raw v_sqrt_f32 (~1 ulp): |x|^4 rel-err ~5e-7, negligible vs f32 accumulation
    float ab = __builtin_amdgcn_sqrtf(s2);
    pw[d][0] = 1.0f; pw[d][1] = ab; pw[d][2] = s2; pw[d][3] = s2 * ab; pw[d][4] = s2 * s2;
    if (d >= 3 && d <= 6) { xvr[6 - d] = re; xvi[6 - d] = im; }
  }

  v8f acc = {};
  gmp_chain<0>(pw, xvr, xvi, ws, half, m, acc);

  // D row 0 (y_re) lives in acc[0], row 1 (y_im) in acc[1], on lanes 0..15
  if (half == 0 && n0 < n_total) {
    out[n0]           = acc[0];
    out[n_total + n0] = acc[1];
  }
}

extern "C" void kernel_launch(void* const* d_in, const int* in_sizes, int n_in,
                              void* d_out, int out_size, void* d_ws, size_t ws_size,
                              hipStream_t stream) {
  const float* x_re = (const float*)d_in[0];
  const float* x_im = (const float*)d_in[1];
  const float* a_re = (const float*)d_in[2];
  const float* a_im = (const float*)d_in[3];
  const float* b_re = (const float*)d_in[4];
  const float* b_im = (const float*)d_in[5];
  const float* c_re = (const float*)d_in[6];
  const float* c_im = (const float*)d_in[7];
  float* out = (float*)d_out;

  int n = in_sizes[0];
  int blocks = (n + 127) / 128;   // 128 samples per 256-thread (8-wave) block
  gmp_wmma_kernel<<<blocks, 256, 0, stream>>>(x_re, x_im, a_re, a_im,
                                              b_re, b_im, c_re, c_im, out, n);
}